// EmbeddingGroup_72456098284169
// MI455X (gfx1250) — compile-verified
//
#include <hip/hip_runtime.h>
#include <math.h>

typedef __attribute__((ext_vector_type(2))) float v2f;
typedef __attribute__((ext_vector_type(8))) float v8f;

#define NE   256
#define ED   1024
#define NPIX 32768
#define HW   1024
#define CHW  (1024*1024)

// d_out layout (floats), reference return order
#define OUT_OFF_LOSS 33554432
#define OUT_OFF_PERP 33554433
#define OUT_OFF_OH   33554434
#define OUT_OFF_IDX  41943042

// workspace layout (float units)
#define WS_EW   0          // 262144 floats: scaled codebook [256][1024]
#define WS_EWN  262144     // 256 floats: ||ew_e||^2
#define WS_SC   262400     // [0]=std, [1]=mean
#define WS_ACC  262402     // loss accumulator
#define WS_IDX  262404     // 32768 ints
#define WS_CNT  295172     // 256 ints (histogram)

#define KC       32        // K-chunk staged in LDS for the distance GEMM
#define LDSTRIDE 34        // 32 + 2 padding: stride-34 -> distinct banks for 16 lanes

// ---------------------------------------------------------------- prep
__global__ void k_prep(const float* mp, const float* sp, const float* nz, float* ws) {
    int t = threadIdx.x;
    int* cnt = (int*)(ws + WS_CNT);
    if (t < NE) cnt[t] = 0;
    if (t == 0) {
        ws[WS_SC + 0] = fabsf(sp[0]) + nz[0];
        ws[WS_SC + 1] = (mp[0] + mp[1] + mp[2]) * (1.0f / 3.0f);
        ws[WS_ACC]    = 0.0f;
    }
}

// --------------------------------------------------- scaled codebook + norms
__global__ void k_ew(const float* __restrict__ emb, float* __restrict__ ws) {
    __shared__ float red[256];
    int e = blockIdx.x, t = threadIdx.x;
    float stdv = ws[WS_SC + 0], meanv = ws[WS_SC + 1];
    float s = 0.0f;
    for (int j = t; j < ED; j += 256) {
        float v = emb[e * ED + j] * stdv + meanv;
        ws[WS_EW + e * ED + j] = v;
        s += v * v;
    }
    red[t] = s; __syncthreads();
    for (int off = 128; off; off >>= 1) { if (t < off) red[t] += red[t + off]; __syncthreads(); }
    if (t == 0) ws[WS_EWN + e] = red[0];
}

// --------------------------------- distance GEMM (WMMA f32) + fused argmin
// Block = 8 waves x 16 pixels each; codebook K-chunk staged in LDS once per
// block via async global->LDS (ASYNCcnt), B-fragments read as ds_load_b64.
__global__ __launch_bounds__(256, 1)
void k_dist(const float* __restrict__ z, float* __restrict__ ws, float* __restrict__ out_idx_f) {
    __shared__ float lds_ew[NE * LDSTRIDE];   // 256 x 34 floats = 34816 B

    const float* ew  = ws + WS_EW;
    const float* ewn = ws + WS_EWN;
    int* idxw = (int*)(ws + WS_IDX);

    const int tid  = threadIdx.x;
    const int lane = tid & 31;
    const int wv   = tid >> 5;
    const int m    = lane & 15;          // A row / B col within tile
    const int half = lane >> 4;
    const int kk_off = half * 2;         // K sub-offset per ISA A/B layout
    const int n0 = (blockIdx.x * 8 + wv) * 16;

    const int n = n0 + m;
    const long long zbase = (long long)(n >> 10) * CHW + (n & 1023);

    v8f acc[16];
    const v8f vzero = {0.f,0.f,0.f,0.f,0.f,0.f,0.f,0.f};
#pragma unroll
    for (int t = 0; t < 16; t++) acc[t] = vzero;

    for (int k0c = 0; k0c < ED; k0c += KC) {
        // cooperative async copy of ew[:, k0c:k0c+KC] -> LDS (2048 b128 xfers)
#pragma unroll
        for (int i = 0; i < 8; i++) {
            int q   = i * 256 + tid;
            int row = q >> 3;            // 8 x b128 per 32-float row
            int c16 = q & 7;
            const float* g = ew + row * ED + k0c + c16 * 4;
            unsigned l = (unsigned)(size_t)(&lds_ew[row * LDSTRIDE + c16 * 4]);
            asm volatile("global_load_async_to_lds_b128 %0, %1, off"
                         :: "v"(l), "v"(g) : "memory");
        }
        asm volatile("s_wait_asynccnt 0x0" ::: "memory");
        __syncthreads();

#pragma unroll
        for (int kk0 = 0; kk0 < KC; kk0 += 4) {
            const int kloc = kk0 + kk_off;            // even -> 8B-aligned ds_load_b64
            v2f a;
            a.x = z[zbase + (long long)(k0c + kloc) * HW];
            a.y = z[zbase + (long long)(k0c + kloc + 1) * HW];
#pragma unroll
            for (int tt = 0; tt < 16; tt++) {
                const int e = tt * 16 + m;
                v2f b = *(const v2f*)&lds_ew[e * LDSTRIDE + kloc];
                acc[tt] = __builtin_amdgcn_wmma_f32_16x16x4_f32(
                    false, a, false, b, (short)0, acc[tt], false, false);
            }
        }
        __syncthreads();   // protect LDS before next chunk's copies
    }

    // score = ||e||^2 - 2 * dot ; argmin over 256 codes (first-index ties)
    float bestv[8]; int bestc[8];
#pragma unroll
    for (int r = 0; r < 8; r++) { bestv[r] = 3.4e38f; bestc[r] = 0; }
#pragma unroll
    for (int t = 0; t < 16; t++) {
        const int   c  = t * 16 + m;
        const float en = ewn[c];
#pragma unroll
        for (int r = 0; r < 8; r++) {
            float v = en - 2.0f * acc[t][r];
            if (v < bestv[r] || (v == bestv[r] && c < bestc[r])) { bestv[r] = v; bestc[r] = c; }
        }
    }
#pragma unroll
    for (int r = 0; r < 8; r++) {
#pragma unroll
        for (int msk = 8; msk >= 1; msk >>= 1) {   // stays within each 16-lane group
            float ov = __shfl_xor(bestv[r], msk, 32);
            int   oc = __shfl_xor(bestc[r], msk, 32);
            if (ov < bestv[r] || (ov == bestv[r] && oc < bestc[r])) { bestv[r] = ov; bestc[r] = oc; }
        }
        if (m == 0) {
            int row = n0 + r + half * 8;
            idxw[row] = bestc[r];
            out_idx_f[row] = (float)bestc[r];
        }
    }
}

// ---------------------------------------------------------------- one-hot
__global__ void k_onehot(const float* __restrict__ ws, float* __restrict__ oh) {
    const int* idxw = (const int*)(ws + WS_IDX);
    int j = blockIdx.x * 256 + threadIdx.x;     // [N, 256] row-major
    oh[j] = (idxw[j >> 8] == (j & 255)) ? 1.0f : 0.0f;
}

// --------------------------------------------------------------- histogram
__global__ void k_hist(float* ws) {
    const int* idxw = (const int*)(ws + WS_IDX);
    int* cnt = (int*)(ws + WS_CNT);
    int nn = blockIdx.x * 256 + threadIdx.x;
    atomicAdd(&cnt[idxw[nn]], 1);
}

// ------------------------------------------------------------ loss partial
__global__ void k_loss(const float* __restrict__ z, float* __restrict__ ws) {
    const float* ew = ws + WS_EW;
    const int* idxw = (const int*)(ws + WS_IDX);
    float s = 0.0f;
    const int total = CHW * 32;
    for (int j = blockIdx.x * 256 + threadIdx.x; j < total; j += gridDim.x * 256) {
        int b  = j >> 20;
        int c  = (j >> 10) & 1023;
        int hw = j & 1023;
        int nn = (b << 10) | hw;
        float d = ew[idxw[nn] * ED + c] - z[j];
        s += d * d;
    }
    __shared__ float red[256];
    red[threadIdx.x] = s; __syncthreads();
    for (int off = 128; off; off >>= 1) {
        if (threadIdx.x < off) red[threadIdx.x] += red[threadIdx.x + off];
        __syncthreads();
    }
    if (threadIdx.x == 0) atomicAdd(&ws[WS_ACC], red[0]);
}

// ------------------------- 1x1 conv GEMM (WMMA f32), out^T orientation
// 64x64 register tile per wave: 4 A-frags (conv_w rows o) x 4 B-frags
// (gathered codebook rows n) -> 16 accumulators, 128 B fresh per WMMA.
__global__ __launch_bounds__(256, 1)
void k_conv(const float* __restrict__ cw, const float* __restrict__ cb,
            const float* __restrict__ ws, float* __restrict__ out) {
    const float* ew = ws + WS_EW;
    const int* idxw = (const int*)(ws + WS_IDX);

    const int lane = threadIdx.x & 31;
    const int wv   = threadIdx.x >> 5;
    const int m    = lane & 15;
    const int half = lane >> 4;
    const int kk_off = half * 2;

    const int o0 = blockIdx.x * 64;                  // 64 output channels per wave
    const int n0 = (blockIdx.y * 8 + wv) * 64;       // 64 pixels per wave

    int rowreg[4];
#pragma unroll
    for (int tb = 0; tb < 4; tb++) rowreg[tb] = idxw[n0 + tb * 16 + m];

    v8f acc[16];
    const v8f vzero = {0.f,0.f,0.f,0.f,0.f,0.f,0.f,0.f};
#pragma unroll
    for (int t = 0; t < 16; t++) acc[t] = vzero;

    for (int k0 = 0; k0 < ED; k0 += 4) {
        const int kk = k0 + kk_off;
        v2f a[4];
#pragma unroll
        for (int ta = 0; ta < 4; ta++)
            a[ta] = *(const v2f*)(cw + (o0 + ta * 16 + m) * ED + kk);   // L2-hot
#pragma unroll
        for (int tb = 0; tb < 4; tb++) {
            v2f b = *(const v2f*)(ew + rowreg[tb] * ED + kk);
#pragma unroll
            for (int ta = 0; ta < 4; ta++) {
                acc[ta * 4 + tb] = __builtin_amdgcn_wmma_f32_16x16x4_f32(
                    false, a[ta], false, b, (short)0, acc[ta * 4 + tb], false, false);
            }
        }
    }

#pragma unroll
    for (int ta = 0; ta < 4; ta++) {
#pragma unroll
        for (int r = 0; r < 8; r++) {
            const int o = o0 + ta * 16 + r + half * 8;
            const float bias = cb[o];
#pragma unroll
            for (int tb = 0; tb < 4; tb++) {
                const int n = n0 + tb * 16 + m;      // consecutive across lanes 0..15
                long long off = (long long)(n >> 10) * CHW + (long long)o * HW + (n & 1023);
                out[off] = acc[ta * 4 + tb][r] + bias;
            }
        }
    }
}

// ----------------------------------------------------- loss & perplexity
__global__ void k_final(const float* __restrict__ ws, float* __restrict__ out) {
    __shared__ float red[256];
    const int* cnt = (const int*)(ws + WS_CNT);
    int t = threadIdx.x;
    float e = (float)cnt[t] * (1.0f / 32768.0f);
    red[t] = e * logf(e + 1e-10f);
    __syncthreads();
    for (int off = 128; off; off >>= 1) { if (t < off) red[t] += red[t + off]; __syncthreads(); }
    if (t == 0) {
        out[OUT_OFF_PERP] = expf(-red[0]);
        out[OUT_OFF_LOSS] = ws[WS_ACC] * 1.25f / 33554432.0f;  // (1+beta)*MSE
    }
}

extern "C" void kernel_launch(void* const* d_in, const int* in_sizes, int n_in,
                              void* d_out, int out_size, void* d_ws, size_t ws_size,
                              hipStream_t stream) {
    const float* z   = (const float*)d_in[0];
    const float* emb = (const float*)d_in[1];
    const float* mp  = (const float*)d_in[2];
    const float* sp  = (const float*)d_in[3];
    const float* nz  = (const float*)d_in[4];
    const float* cw  = (const float*)d_in[5];
    const float* cb  = (const float*)d_in[6];
    float* out = (float*)d_out;
    float* ws  = (float*)d_ws;

    k_prep  <<<1, 256, 0, stream>>>(mp, sp, nz, ws);
    k_ew    <<<NE, 256, 0, stream>>>(emb, ws);
    k_dist  <<<NPIX / 128, 256, 0, stream>>>(z, ws, out + OUT_OFF_IDX);
    k_onehot<<<(NPIX * NE) / 256, 256, 0, stream>>>(ws, out + OUT_OFF_OH);
    k_hist  <<<NPIX / 256, 256, 0, stream>>>(ws);
    k_loss  <<<1024, 256, 0, stream>>>(z, ws);
    k_conv  <<<dim3(16, 64), 256, 0, stream>>>(cw, cb, ws, out);
    k_final <<<1, 256, 0, stream>>>(ws, out);
}